// ControlUnit_52321291600660
// MI455X (gfx1250) — compile-verified
//
#include <hip/hip_runtime.h>
#include <math.h>

typedef __attribute__((ext_vector_type(2))) float v2f;
typedef __attribute__((ext_vector_type(8))) float v8f;

#define BB   64
#define LL   512
#define DIMD 2048
#define KH   2048          // half of concat K (control | question)
#define NCH  8             // L chunks per batch for split flash-softmax
#define LCH  (LL / NCH)    // 64 positions per chunk

// ---------------------------------------------------------------------------
// Kernel 1: v[b,n] = ( [ctrl|quest] @ W_cq + b_cq + step_emb[step] )[b,n] * W_attn[n]
// One wave per 16-wide N tile; wave computes all 4 M tiles (M=64) so W_cq is
// streamed exactly once from HBM. Uses V_WMMA_F32_16X16X4_F32.
// ---------------------------------------------------------------------------
__global__ void __launch_bounds__(256) cq_proj_wmma(
    const float* __restrict__ ctrl, const float* __restrict__ quest,
    const float* __restrict__ W,    const float* __restrict__ b_cq,
    const float* __restrict__ step_emb, const int* __restrict__ step,
    const float* __restrict__ W_attn,   float* __restrict__ v_out)
{
    const int wid  = threadIdx.x >> 5;
    const int lane = threadIdx.x & 31;
    const int n_tile = blockIdx.x * 8 + wid;        // 0..127
    const int n0   = n_tile * 16;
    const int row  = lane & 15;                      // M (A) / N (B,C) index
    const int half = lane >> 4;                      // K-half selector

    const float* A0 = ctrl  + (size_t)row * KH;      // +16/32/48 rows per frag
    const float* A1 = quest + (size_t)row * KH;

    v8f acc0 = {}, acc1 = {}, acc2 = {}, acc3 = {};

    // ---- K first half: control_state ----
    #pragma unroll 2
    for (int k = 0; k < KH; k += 4) {
        const int kk = k + half * 2;
        v2f bf;
        bf.x = W[(size_t)kk * DIMD + n0 + row];
        bf.y = W[(size_t)(kk + 1) * DIMD + n0 + row];
        v2f a0 = *(const v2f*)(A0 + kk);
        v2f a1 = *(const v2f*)(A0 + (size_t)16 * KH + kk);
        v2f a2 = *(const v2f*)(A0 + (size_t)32 * KH + kk);
        v2f a3 = *(const v2f*)(A0 + (size_t)48 * KH + kk);
        acc0 = __builtin_amdgcn_wmma_f32_16x16x4_f32(false, a0, false, bf, (short)0, acc0, false, false);
        acc1 = __builtin_amdgcn_wmma_f32_16x16x4_f32(false, a1, false, bf, (short)0, acc1, false, false);
        acc2 = __builtin_amdgcn_wmma_f32_16x16x4_f32(false, a2, false, bf, (short)0, acc2, false, false);
        acc3 = __builtin_amdgcn_wmma_f32_16x16x4_f32(false, a3, false, bf, (short)0, acc3, false, false);
    }
    // ---- K second half: question (W rows offset by KH) ----
    const float* W2 = W + (size_t)KH * DIMD;
    #pragma unroll 2
    for (int k = 0; k < KH; k += 4) {
        const int kk = k + half * 2;
        v2f bf;
        bf.x = W2[(size_t)kk * DIMD + n0 + row];
        bf.y = W2[(size_t)(kk + 1) * DIMD + n0 + row];
        v2f a0 = *(const v2f*)(A1 + kk);
        v2f a1 = *(const v2f*)(A1 + (size_t)16 * KH + kk);
        v2f a2 = *(const v2f*)(A1 + (size_t)32 * KH + kk);
        v2f a3 = *(const v2f*)(A1 + (size_t)48 * KH + kk);
        acc0 = __builtin_amdgcn_wmma_f32_16x16x4_f32(false, a0, false, bf, (short)0, acc0, false, false);
        acc1 = __builtin_amdgcn_wmma_f32_16x16x4_f32(false, a1, false, bf, (short)0, acc1, false, false);
        acc2 = __builtin_amdgcn_wmma_f32_16x16x4_f32(false, a2, false, bf, (short)0, acc2, false, false);
        acc3 = __builtin_amdgcn_wmma_f32_16x16x4_f32(false, a3, false, bf, (short)0, acc3, false, false);
    }

    const int st = step[0];
    const int n  = n0 + row;
    const float add = b_cq[n] + step_emb[(size_t)st * DIMD + n];
    const float wa  = W_attn[n];
    v8f accs[4] = {acc0, acc1, acc2, acc3};
    #pragma unroll
    for (int t = 0; t < 4; ++t)
        #pragma unroll
        for (int i = 0; i < 8; ++i) {
            const int m = t * 16 + i + 8 * half;          // C layout: VGPR i -> M=i / M=8+i
            v_out[(size_t)m * DIMD + n] = (accs[t][i] + add) * wa;
        }
}

// ---------------------------------------------------------------------------
// Kernel 2: flash-style single pass over context. Block = (b, chunk of 64 L).
// Each wave owns 8 rows sequentially: one read of ctx row feeds both the logit
// dot-product and the online-softmax weighted accumulation.
// ---------------------------------------------------------------------------
__global__ void __launch_bounds__(256) attn_partial(
    const float* __restrict__ context, const unsigned char* __restrict__ q_mask,
    const float* __restrict__ b_attn,  const float* __restrict__ v,
    float* __restrict__ partial)
{
    __shared__ float v_s[DIMD];
    __shared__ float acc_s[DIMD];
    __shared__ float m_s[8];
    __shared__ float s_s[8];

    const int b    = blockIdx.x;
    const int ch   = blockIdx.y;
    const int tid  = threadIdx.x;
    const int wid  = tid >> 5;
    const int lane = tid & 31;

    for (int i = tid; i < DIMD; i += 256) {
        v_s[i]   = v[(size_t)b * DIMD + i];
        acc_s[i] = 0.0f;
    }
    __syncthreads();

    const float ba = b_attn[0];
    float m = -INFINITY, s = 0.0f;
    float4 acc[16];
    #pragma unroll
    for (int j = 0; j < 16; ++j) acc[j] = make_float4(0.f, 0.f, 0.f, 0.f);

    const int l0 = ch * LCH + wid * (LCH / 8);       // 8 rows per wave
    for (int li = 0; li < LCH / 8; ++li) {
        const int l = l0 + li;
        const float* ctx = context + ((size_t)b * LL + l) * DIMD;
        float4 c[16];
        float dot = 0.0f;
        #pragma unroll
        for (int j = 0; j < 16; ++j) {
            const int idx = lane * 4 + j * 128;
            c[j] = *(const float4*)(ctx + idx);
            const float4 vv = *(const float4*)(v_s + idx);
            dot += c[j].x * vv.x + c[j].y * vv.y + c[j].z * vv.z + c[j].w * vv.w;
        }
        #pragma unroll
        for (int off = 16; off > 0; off >>= 1)
            dot += __shfl_xor(dot, off, 32);

        float logit = dot + ba;
        if (!q_mask[(size_t)b * LL + l]) logit = -10000.0f;

        const float mn    = fmaxf(m, logit);
        const float scale = __expf(m - mn);          // first iter: exp(-inf)=0
        const float w     = __expf(logit - mn);
        s = s * scale + w;
        #pragma unroll
        for (int j = 0; j < 16; ++j) {
            acc[j].x = acc[j].x * scale + w * c[j].x;
            acc[j].y = acc[j].y * scale + w * c[j].y;
            acc[j].z = acc[j].z * scale + w * c[j].z;
            acc[j].w = acc[j].w * scale + w * c[j].w;
        }
        m = mn;
    }

    if (lane == 0) { m_s[wid] = m; s_s[wid] = s; }
    __syncthreads();

    float mstar = m_s[0];
    #pragma unroll
    for (int wv = 1; wv < 8; ++wv) mstar = fmaxf(mstar, m_s[wv]);
    const float scw = __expf(m - mstar);

    #pragma unroll
    for (int j = 0; j < 16; ++j) {
        const int idx = lane * 4 + j * 128;
        atomicAdd(&acc_s[idx + 0], scw * acc[j].x);
        atomicAdd(&acc_s[idx + 1], scw * acc[j].y);
        atomicAdd(&acc_s[idx + 2], scw * acc[j].z);
        atomicAdd(&acc_s[idx + 3], scw * acc[j].w);
    }
    __syncthreads();

    float* P = partial + ((size_t)b * NCH + ch) * (DIMD + 2);
    if (tid == 0) {
        float st = 0.0f;
        for (int wv = 0; wv < 8; ++wv) st += __expf(m_s[wv] - mstar) * s_s[wv];
        P[0] = mstar;
        P[1] = st;
    }
    for (int i = tid; i < DIMD; i += 256) P[2 + i] = acc_s[i];
}

// ---------------------------------------------------------------------------
// Kernel 3: merge NCH partial (m, s, acc) states per batch -> new_control.
// ---------------------------------------------------------------------------
__global__ void __launch_bounds__(256) attn_finalize(
    const float* __restrict__ partial, float* __restrict__ out)
{
    const int b   = blockIdx.x;
    const int tid = threadIdx.x;
    const float* Pb = partial + (size_t)b * NCH * (DIMD + 2);

    float mstar = -INFINITY;
    #pragma unroll
    for (int c = 0; c < NCH; ++c) mstar = fmaxf(mstar, Pb[(size_t)c * (DIMD + 2)]);

    float coef[NCH];
    float S = 0.0f;
    #pragma unroll
    for (int c = 0; c < NCH; ++c) {
        coef[c] = __expf(Pb[(size_t)c * (DIMD + 2)] - mstar);
        S += coef[c] * Pb[(size_t)c * (DIMD + 2) + 1];
    }
    const float inv = 1.0f / S;

    for (int d = tid; d < DIMD; d += 256) {
        float num = 0.0f;
        #pragma unroll
        for (int c = 0; c < NCH; ++c)
            num += coef[c] * Pb[(size_t)c * (DIMD + 2) + 2 + d];
        out[(size_t)b * DIMD + d] = num * inv;
    }
}

// ---------------------------------------------------------------------------
extern "C" void kernel_launch(void* const* d_in, const int* in_sizes, int n_in,
                              void* d_out, int out_size, void* d_ws, size_t ws_size,
                              hipStream_t stream) {
    const int*           step      = (const int*)d_in[0];
    const float*         context   = (const float*)d_in[1];
    const float*         question  = (const float*)d_in[2];
    const float*         ctrl      = (const float*)d_in[3];
    const unsigned char* q_mask    = (const unsigned char*)d_in[4];
    const float*         W_cq      = (const float*)d_in[5];
    const float*         b_cq      = (const float*)d_in[6];
    const float*         step_emb  = (const float*)d_in[7];
    const float*         W_attn    = (const float*)d_in[8];
    const float*         b_attn    = (const float*)d_in[9];
    float* out = (float*)d_out;

    float* v       = (float*)d_ws;                       // B*DIM floats (512 KB)
    float* partial = v + (size_t)BB * DIMD;              // B*NCH*(DIM+2) floats (~4.2 MB)

    // 128 N-tiles, 8 waves/block -> 16 blocks
    cq_proj_wmma<<<16, 256, 0, stream>>>(ctrl, question, W_cq, b_cq, step_emb,
                                         step, W_attn, v);
    attn_partial<<<dim3(BB, NCH), 256, 0, stream>>>(context, q_mask, b_attn, v, partial);
    attn_finalize<<<BB, 256, 0, stream>>>(partial, out);
}